// DelayNoSumGather_37890201485326
// MI455X (gfx1250) — compile-verified
//
#include <hip/hip_runtime.h>
#include <stdint.h>

// Problem constants (match reference)
#define FSf     4.0e7f
#define Bb      2
#define Tt      2030
#define Ee      256
#define Yy      256
#define Xx      256

// Tiling
#define ETILE   8      // e-values per workgroup (== waves per WG)
#define YCHUNK  16     // y rows per workgroup
#define SPITCH  9      // dwords per t-row in LDS (8 data + 1 TDM pad) -> bank spread
#define DPITCH  260    // dwords per e-row of the distance transpose tile (260 % 64 == 4)

typedef unsigned int u32;
typedef u32 v4u __attribute__((ext_vector_type(4)));
typedef int  v8i __attribute__((ext_vector_type(8)));
typedef int  v4i __attribute__((ext_vector_type(4)));

__global__ __launch_bounds__(256)
void delay_nosum_gather(const float* __restrict__ sino,   // [B,T,E]
                        const float* __restrict__ sos,    // [B]
                        const float* __restrict__ dist,   // [Y,X,E]
                        float* __restrict__ out)          // [B,E,Y,X]
{
    extern __shared__ float smem[];
    float* sLDS = smem;                     // sinogram columns, [Tt][SPITCH]
    float* dLDS = smem + Tt * SPITCH;       // distance transpose tile, [ETILE][DPITCH]

    const int tid  = threadIdx.x;
    const u32 wg   = blockIdx.x;
    const int b    = (int)(wg & 1u);
    const int et   = (int)((wg >> 1) & 31u);
    const int yc   = (int)(wg >> 6);
    const int e0   = et * ETILE;
    const int y0   = yc * YCHUNK;

    // ---------------- Stage sinogram columns sino[b, :, e0..e0+7] into LDS ----------------
#if defined(__has_builtin) && __has_builtin(__builtin_amdgcn_tensor_load_to_lds)
    if (tid < 32) {   // one wave issues the TDM descriptor (wave-level op)
        const float* gsrc = sino + ((size_t)b * Tt) * Ee + e0;  // tile start in tensor
        uint64_t ga = (uint64_t)(uintptr_t)gsrc;
        u32 ldsoff  = (u32)(uintptr_t)sLDS;  // flat LDS pointer: low 32 bits = LDS byte addr

        // D# group 0 (128b): count=1 valid, lds_addr, global_addr[56:0], type=2
        v4u g0;
        g0[0] = 1u;
        g0[1] = ldsoff;
        g0[2] = (u32)(ga & 0xffffffffu);
        g0[3] = (u32)((ga >> 32) & 0x01ffffffu) | 0x80000000u;   // addr[56:32] | type=2<<30

        // D# group 1 (256b)
        v8i g1;
        // data_size=4B (code 2), pad_enable=1, pad_interval=8 dwords (code 2), pad_amount=1 dword (code 0)
        g1[0] = (int)((2u << 16) | (1u << 20) | (2u << 22));
        g1[1] = (int)((u32)ETILE << 16);         // tensor_dim0 lo16 (=8); barrier addr = 0
        g1[2] = (int)((u32)Tt << 16);            // tensor_dim0 hi16 (=0) | tensor_dim1 lo16 (=2030)
        g1[3] = (int)((u32)ETILE << 16);         // tensor_dim1 hi16 (=0) | tile_dim0 (=8)
        g1[4] = (int)((u32)Tt & 0xffffu);        // tile_dim1 (=2030) | tile_dim2 (=0)
        g1[5] = (int)Ee;                         // tensor_dim0_stride lo32 (=256 elements)
        g1[6] = 0;                               // stride0 hi16 | tensor_dim1_stride lo16
        g1[7] = 0;                               // tensor_dim1_stride hi32

        v4i gz4 = {0, 0, 0, 0};                  // 2D tensor: D# groups 2/3 unused
        v8i gz8 = {0, 0, 0, 0, 0, 0, 0, 0};      // trailing group (clang-23 6-arg form)
        __builtin_amdgcn_tensor_load_to_lds(g0, g1, gz4, gz4, gz8, 0);
        __builtin_amdgcn_s_wait_tensorcnt(0);
    }
#else
    // Fallback staging: strided global loads -> padded LDS layout [t][SPITCH]
    for (int idx = tid; idx < Tt * ETILE; idx += 256) {
        int t  = idx >> 3;
        int el = idx & 7;
        sLDS[t * SPITCH + el] = sino[((size_t)b * Tt + t) * Ee + e0 + el];
    }
#endif
    __syncthreads();

    const float scale = FSf / sos[b];     // uniform; tof = d * (FS/sos)
    const int   w     = tid >> 5;         // wave id == local e index
    const int   lane  = tid & 31;
    const int   e     = e0 + w;

    for (int yy = 0; yy < YCHUNK; ++yy) {
        const int y = y0 + yy;

        __syncthreads();   // previous iteration's dLDS reads done
        // Coalesced load of dist[y, 0..255, e0..e0+7], transposed into dLDS[el][x]
        const float* drow = dist + ((size_t)y * Xx) * Ee + e0;
        #pragma unroll
        for (int j = 0; j < 8; ++j) {
            int idx = j * 256 + tid;
            int el  = idx & 7;          // consecutive lanes -> consecutive e (32B chunks)
            int x   = idx >> 3;
            dLDS[el * DPITCH + x] = drow[(size_t)x * Ee + el];
        }
        __syncthreads();

        float* orow = out + (((size_t)b * Ee + e) * Yy + y) * Xx;
        #pragma unroll
        for (int k = 0; k < 8; ++k) {
            int   x   = k * 32 + lane;
            float d   = dLDS[w * DPITCH + x];
            float tof = d * scale;
            float tf  = floorf(tof);
            float fr  = tof - tf;                 // ceil factor; (1-fr) = floor factor
            int i0 = (int)tf;
            i0 = i0 < 0 ? 0 : (i0 > (Tt - 1) ? (Tt - 1) : i0);
            int i1 = (i0 + 1 > (Tt - 1)) ? (Tt - 1) : (i0 + 1);
            float gf = sLDS[i0 * SPITCH + w];     // LDS gather, bank-spread via pitch 9
            float gc = sLDS[i1 * SPITCH + w];
            float v  = (1.0f - fr) * gf + fr * gc;
            __builtin_nontemporal_store(v, orow + x);  // write-once stream; keep L2 for dist
        }
    }
}

extern "C" void kernel_launch(void* const* d_in, const int* in_sizes, int n_in,
                              void* d_out, int out_size, void* d_ws, size_t ws_size,
                              hipStream_t stream) {
    (void)in_sizes; (void)n_in; (void)out_size; (void)d_ws; (void)ws_size;
    const float* sino = (const float*)d_in[0];
    const float* sos  = (const float*)d_in[1];
    const float* dist = (const float*)d_in[2];
    float* out        = (float*)d_out;

    const int nwg = Bb * (Ee / ETILE) * (Yy / YCHUNK);   // 2*32*16 = 1024
    const size_t shmem = (size_t)(Tt * SPITCH + ETILE * DPITCH) * sizeof(float); // ~81.4 KB
    delay_nosum_gather<<<dim3(nwg), dim3(256), shmem, stream>>>(sino, sos, dist, out);
}